// TransformerEncoderLayer_2388001817176
// MI455X (gfx1250) — compile-verified
//
#include <hip/hip_runtime.h>
#include <math.h>

// ---------------------------------------------------------------------------
// TransformerEncoderLayer for MI455X (gfx1250), fp32 end-to-end.
// B=8, A=128, D=256, H=8, DFF=2048.
//  * k_pos/v_pos projections (rel @ Wkp/Wvp, 2x17 GFLOP over a 134MB tensor)
//    are fused into the attention kernel: rel[b,q] (128KB) is staged once
//    into the 320KB WGP LDS via GLOBAL_LOAD_ASYNC_TO_LDS_B128, projected with
//    v_wmma_f32_16x16x4_f32 into LDS, and consumed in place. k_pos/v_pos
//    never touch HBM (saves ~536MB of traffic @ 23.3 TB/s).
//  * All GEMMs use the native fp32 WMMA op => reference precision.
// ---------------------------------------------------------------------------

typedef __attribute__((ext_vector_type(2))) float v2f;
typedef __attribute__((ext_vector_type(8))) float v8f;

#define D_MODEL 256
#define SEQ     128
#define NH      8

__device__ __forceinline__ v8f wmma4(v2f a, v2f b, v8f c) {
  // (neg_a, A, neg_b, B, c_mod, C, reuse_a, reuse_b)
  return __builtin_amdgcn_wmma_f32_16x16x4_f32(false, a, false, b, (short)0, c,
                                               false, false);
}

// ---------------------------------------------------------------------------
// Generic GEMM:  C[M,N] = A[M,K] @ B[K,N] + bias[N]  (optional ReLU)
// One 16x16 tile per wave32; 8 waves / 256 threads per block.
// A 16x4 fragment: lanes 0-15 -> K={k0,k0+1}, lanes 16-31 -> K={k0+2,k0+3}.
// B 4x16 fragment mirrored.  C/D: VGPR v = row (v + 8*(lane>=16)), col lane&15.
// ---------------------------------------------------------------------------
__global__ void gemm_bias_kernel(const float* __restrict__ A,
                                 const float* __restrict__ B,
                                 const float* __restrict__ bias,
                                 float* __restrict__ C,
                                 int M, int N, int K, int relu) {
  const int wv   = threadIdx.x >> 5;
  const int lane = threadIdx.x & 31;
  const int tilesN = N >> 4;
  const int tiles  = (M >> 4) * tilesN;
  const int wid = blockIdx.x * 8 + wv;
  if (wid >= tiles) return;                       // wave-uniform
  const int m0 = (wid / tilesN) << 4;
  const int n0 = (wid % tilesN) << 4;
  const int r  = lane & 15;
  const int ks = (lane >> 4) << 1;                // 0 or 2
  const int ro = (lane >> 4) << 3;                // 0 or 8

  // strength-reduced pointers: only adds in the K loop
  const float* Ap = A + (size_t)(m0 + r) * K + ks;
  const float* Bp = B + (size_t)ks * N + n0 + r;
  const size_t bstep = (size_t)4 * N;
  v8f acc = {0.f, 0.f, 0.f, 0.f, 0.f, 0.f, 0.f, 0.f};
  for (int k0 = 0; k0 < K; k0 += 4) {
    v2f a, b;
    a.x = Ap[0];
    a.y = Ap[1];
    b.x = Bp[0];
    b.y = Bp[(size_t)N];
    acc = wmma4(a, b, acc);
    Ap += 4;
    Bp += bstep;
  }
  const float bi = bias[n0 + r];
  float* Crow = C + (size_t)(m0 + ro) * N + n0 + r;
#pragma unroll
  for (int v = 0; v < 8; ++v) {
    float val = acc[v] + bi;
    if (relu) val = fmaxf(val, 0.f);
    Crow[(size_t)v * N] = val;
  }
}

// ---------------------------------------------------------------------------
// LDS-sourced GEMM used inside the fused attention kernel:
// out_s[128,256] = rel_s[128,256] @ W[256,256] + bias.   8 waves, 128 tiles.
// ---------------------------------------------------------------------------
__device__ __forceinline__ void gemm_rel_lds(const float* rel_s,
                                             const float* __restrict__ W,
                                             const float* __restrict__ bias,
                                             float* out_s, int wv, int lane) {
  const int r  = lane & 15;
  const int ks = (lane >> 4) << 1;
  const int ro = (lane >> 4) << 3;
  for (int tile = wv; tile < (SEQ / 16) * (D_MODEL / 16); tile += 8) {
    const int m0 = (tile >> 4) << 4;              // 8 row tiles
    const int n0 = (tile & 15) << 4;              // 16 col tiles
    const float* As = rel_s + (m0 + r) * D_MODEL + ks;  // LDS (ds_load_b64)
    const float* Bp = W + (size_t)ks * D_MODEL + n0 + r;
    v8f acc = {0.f, 0.f, 0.f, 0.f, 0.f, 0.f, 0.f, 0.f};
#pragma unroll 4
    for (int k0 = 0; k0 < D_MODEL; k0 += 4) {
      v2f a, b;
      a.x = As[0];
      a.y = As[1];
      b.x = Bp[0];
      b.y = Bp[D_MODEL];
      acc = wmma4(a, b, acc);
      As += 4;
      Bp += 4 * D_MODEL;
    }
    const float bi = bias[n0 + r];
    float* Co = out_s + (m0 + ro) * D_MODEL + n0 + r;
#pragma unroll
    for (int v = 0; v < 8; ++v) Co[v * D_MODEL] = acc[v] + bi;
  }
}

// Block-wide sum over 256 threads (8 wave32s), via shuffles + tiny LDS.
__device__ __forceinline__ float block_sum_256(float v, float* red) {
  const int lane = threadIdx.x & 31;
  const int wv   = threadIdx.x >> 5;
#pragma unroll
  for (int off = 16; off > 0; off >>= 1) v += __shfl_xor(v, off, 32);
  if (lane == 0) red[wv] = v;
  __syncthreads();
  if (wv == 0) {
    float s = (lane < 8) ? red[lane] : 0.f;
#pragma unroll
    for (int off = 16; off > 0; off >>= 1) s += __shfl_xor(s, off, 32);
    if (lane == 0) red[8] = s;
  }
  __syncthreads();
  float s = red[8];
  __syncthreads();                                // allow red reuse
  return s;
}

// ---------------------------------------------------------------------------
// Fused attention: one workgroup per (b,q).  ~262KB dynamic LDS (CDNA5 WGP).
//   rel_s : rel[b,q]            128x256  (128KB)  <- async-LDS staged
//   kv_s  : kpos, later vpos    128x256  (128KB)
// ---------------------------------------------------------------------------
__global__ void attn_fused_kernel(const float* __restrict__ src,
                                  const float* __restrict__ rel,
                                  const float* __restrict__ qg,
                                  const float* __restrict__ qpg,
                                  const float* __restrict__ k_base,
                                  const float* __restrict__ v_base,
                                  const float* __restrict__ Wkp,
                                  const float* __restrict__ bkp,
                                  const float* __restrict__ Wvp,
                                  const float* __restrict__ bvp,
                                  const float* __restrict__ g1,
                                  const float* __restrict__ be1,
                                  float* __restrict__ xout) {
  extern __shared__ float smem[];
  float* rel_s = smem;                         // 32768
  float* kv_s  = rel_s + SEQ * D_MODEL;        // 32768
  float* q_s   = kv_s + SEQ * D_MODEL;         // 256
  float* qp_s  = q_s + D_MODEL;                // 256
  float* sc_s  = qp_s + D_MODEL;               // 8*128 scores -> attn
  float* red_s = sc_s + NH * SEQ;              // 16

  const int bq   = blockIdx.x;                 // b*128 + q
  const int b    = bq >> 7;
  const int t    = threadIdx.x;
  const int wv   = t >> 5;
  const int lane = t & 31;

  // ---- stage rel[b,q] (128KB) directly into LDS via async copy -----------
  // GLOBAL_LOAD_ASYNC_TO_LDS_B128, GVS mode: SGPR base + per-lane i32 offset;
  // LDS destination address carried in the VDST VGPR. Tracked by ASYNCcnt.
  {
    const unsigned long long gbase =
        (unsigned long long)(const void*)(rel + (size_t)bq * SEQ * D_MODEL);
    const unsigned ldsbase = (unsigned)(size_t)(void*)rel_s;
    for (int i = t; i < SEQ * D_MODEL / 4; i += 256) {
      const unsigned goff  = (unsigned)i * 16u;   // bytes
      const unsigned laddr = ldsbase + (unsigned)i * 16u;
      asm volatile("global_load_async_to_lds_b128 %0, %1, %2 offset:0"
                   :
                   : "v"(laddr), "v"(goff), "s"(gbase)
                   : "memory");
    }
    q_s[t]  = qg[(size_t)bq * D_MODEL + t];
    qp_s[t] = qpg[(size_t)bq * D_MODEL + t];
    asm volatile("s_wait_asynccnt 0x0" ::: "memory");
  }
  __syncthreads();

  // ---- kpos = rel @ Wkp + bkp  (WMMA, LDS->LDS) --------------------------
  gemm_rel_lds(rel_s, Wkp, bkp, kv_s, wv, lane);
  __syncthreads();

  // ---- scores: heads 0-3 content (q . k_base), heads 4-7 pos (qp . kpos) -
  {
    const int kk = t & 127;
    const int hb = t >> 7;                      // 0 or 1
#pragma unroll
    for (int hh = 0; hh < 4; ++hh) {
      const int h = hb + 2 * hh;                // covers 0..7
      float s = 0.f;
      if (h < 4) {
        const float* kc = k_base + ((size_t)b * SEQ + kk) * D_MODEL + h * 64;
        const float* qr = q_s + h * 64;
        for (int e = 0; e < 64; ++e) s += qr[e] * kc[e];
      } else {
        const float* kp = kv_s + kk * D_MODEL + (h - 4) * 64;
        const float* qr = qp_s + (h - 4) * 64;
        for (int e = 0; e < 64; ++e) s += qr[e] * kp[e];
      }
      sc_s[h * SEQ + kk] = s * 0.125f;          // 1/sqrt(64)
    }
  }
  __syncthreads();

  // ---- softmax over k, one wave32 per head -------------------------------
  {
    float* row = sc_s + wv * SEQ;
    float vals[4];
    float m = -1e30f;
#pragma unroll
    for (int j = 0; j < 4; ++j) {
      vals[j] = row[lane + 32 * j];
      m = fmaxf(m, vals[j]);
    }
#pragma unroll
    for (int off = 16; off > 0; off >>= 1) m = fmaxf(m, __shfl_xor(m, off, 32));
    float sum = 0.f;
#pragma unroll
    for (int j = 0; j < 4; ++j) { vals[j] = __expf(vals[j] - m); sum += vals[j]; }
#pragma unroll
    for (int off = 16; off > 0; off >>= 1) sum += __shfl_xor(sum, off, 32);
    const float inv = 1.f / sum;
#pragma unroll
    for (int j = 0; j < 4; ++j) row[lane + 32 * j] = vals[j] * inv;
  }
  __syncthreads();

  // ---- vpos = rel @ Wvp + bvp (WMMA), overwrites kpos in LDS -------------
  gemm_rel_lds(rel_s, Wvp, bvp, kv_s, wv, lane);
  __syncthreads();

  // ---- out = attn @ (v_base + vpos), then residual + LayerNorm1 ----------
  {
    const int h = t >> 5;                       // head
    const int e = t & 31;                       // value dim within head
    const float* ar  = sc_s + h * SEQ;
    const float* vbc = v_base + (size_t)b * SEQ * D_MODEL + h * 32 + e;
    const float* vpc = kv_s + h * 32 + e;
    float o = 0.f;
    for (int k = 0; k < SEQ; ++k) {
      o += ar[k] * (vbc[0] + vpc[0]);
      vbc += D_MODEL;
      vpc += D_MODEL;
    }

    float xin = src[(size_t)bq * D_MODEL + t] + o;
    const float mean = block_sum_256(xin, red_s) * (1.f / D_MODEL);
    const float d    = xin - mean;
    const float var  = block_sum_256(d * d, red_s) * (1.f / D_MODEL);
    xout[(size_t)bq * D_MODEL + t] =
        d * rsqrtf(var + 1e-5f) * g1[t] + be1[t];
  }
}

// out = LayerNorm(a + c) with gain/bias; one 256-thread block per row.
__global__ void add_ln_kernel(const float* __restrict__ a,
                              const float* __restrict__ c,
                              const float* __restrict__ g,
                              const float* __restrict__ be,
                              float* __restrict__ out) {
  __shared__ float red[16];
  const int row = blockIdx.x, t = threadIdx.x;
  float x = a[(size_t)row * D_MODEL + t] + c[(size_t)row * D_MODEL + t];
  const float mean = block_sum_256(x, red) * (1.f / D_MODEL);
  const float d    = x - mean;
  const float var  = block_sum_256(d * d, red) * (1.f / D_MODEL);
  out[(size_t)row * D_MODEL + t] = d * rsqrtf(var + 1e-5f) * g[t] + be[t];
}

// ---------------------------------------------------------------------------
extern "C" void kernel_launch(void* const* d_in, const int* in_sizes, int n_in,
                              void* d_out, int out_size, void* d_ws,
                              size_t ws_size, hipStream_t stream) {
  const float* src = (const float*)d_in[0];
  const float* pos = (const float*)d_in[1];
  const float* rel = (const float*)d_in[2];
  const float* Wq  = (const float*)d_in[3];  const float* bq  = (const float*)d_in[4];
  const float* Wk  = (const float*)d_in[5];  const float* bk  = (const float*)d_in[6];
  const float* Wv  = (const float*)d_in[7];  const float* bv  = (const float*)d_in[8];
  const float* Wqp = (const float*)d_in[9];  const float* bqp = (const float*)d_in[10];
  const float* Wkp = (const float*)d_in[11]; const float* bkp = (const float*)d_in[12];
  const float* Wvp = (const float*)d_in[13]; const float* bvp = (const float*)d_in[14];
  const float* W1  = (const float*)d_in[15]; const float* b1  = (const float*)d_in[16];
  const float* W2  = (const float*)d_in[17]; const float* b2  = (const float*)d_in[18];
  const float* g1  = (const float*)d_in[19]; const float* be1 = (const float*)d_in[20];
  const float* g2  = (const float*)d_in[21]; const float* be2 = (const float*)d_in[22];

  const int M = 8 * 128;                       // B*A rows
  float* ws  = (float*)d_ws;
  float* q   = ws;                             // 1024*256
  float* qp  = q   + M * D_MODEL;
  float* kb  = qp  + M * D_MODEL;
  float* vb  = kb  + M * D_MODEL;
  float* x   = vb  + M * D_MODEL;
  float* ffh = x   + M * D_MODEL;              // 1024*2048
  float* ff  = ffh + M * 2048;

  auto gemm = [&](const float* A, const float* B, const float* bias, float* C,
                  int m, int n, int k, int relu) {
    const int tiles = (m / 16) * (n / 16);
    gemm_bias_kernel<<<(tiles + 7) / 8, 256, 0, stream>>>(A, B, bias, C, m, n,
                                                          k, relu);
  };

  // content / positional projections (WMMA fp32)
  gemm(src, Wq,  bq,  q,  M, D_MODEL, D_MODEL, 0);
  gemm(pos, Wqp, bqp, qp, M, D_MODEL, D_MODEL, 0);
  gemm(src, Wk,  bk,  kb, M, D_MODEL, D_MODEL, 0);
  gemm(kb,  Wv,  bv,  vb, M, D_MODEL, D_MODEL, 0);   // v = Wv(Wk(src))

  // fused rel-projection + attention + LN1  (one WG per (b,q), ~262KB LDS)
  const size_t smem =
      (2 * SEQ * D_MODEL + 2 * D_MODEL + NH * SEQ + 16) * sizeof(float);
  attn_fused_kernel<<<M, 256, smem, stream>>>(src, rel, q, qp, kb, vb, Wkp,
                                              bkp, Wvp, bvp, g1, be1, x);

  // FFN + LN2
  gemm(x,   W1, b1, ffh, M, 2048,    D_MODEL, 1);    // ReLU
  gemm(ffh, W2, b2, ff,  M, D_MODEL, 2048,    0);
  add_ln_kernel<<<M, 256, 0, stream>>>(x, ff, g2, be2, (float*)d_out);
}